// SEQSCD_12094627906328
// MI455X (gfx1250) — compile-verified
//
#include <hip/hip_runtime.h>
#include <hip/hip_bf16.h>
#include <math.h>

#define B_ 64
#define T_ 2048
#define H_ 256
#define H2_ 512
#define MAXH 10
#define THRESH 1.0f
#define RELU_MAX_ 1.0f
#define NORM_SCALAR_ 1.0f

typedef __attribute__((ext_vector_type(2))) float v2f;
typedef __attribute__((ext_vector_type(8))) float v8f;

#if defined(__has_builtin)
#if __has_builtin(__builtin_amdgcn_wmma_f32_16x16x4_f32)
#define HAVE_WMMA_F32 1
#endif
#endif

// ---------------------------------------------------------------- zero output
__global__ void k_zero(float* out, long n) {
  long i = (long)blockIdx.x * blockDim.x + threadIdx.x;
  long stride = (long)gridDim.x * blockDim.x;
  for (; i < n; i += stride) out[i] = 0.f;
}

// ------------------------------------------------- fold W2@W1 into one vector
// v[i] = sum_o W2[0,o] * W1[o,i];  v1p = v[0:256]+v[256:512]; v2 = v[256:512]
// c = W2[0,:]·b1 + b2.  Also build transposed dense B table for the WMMA kernel:
// bmatT[n*256 + k] = (n==0 ? v1p[k] : n==1 ? v2[k] : 0)   (16 KB, L0-resident)
__global__ void k_fold(const float* __restrict__ W1, const float* __restrict__ b1,
                       const float* __restrict__ W2, const float* __restrict__ b2,
                       float* __restrict__ v1p, float* __restrict__ v2v,
                       float* __restrict__ cb, float* __restrict__ bmatT) {
  __shared__ float v[H2_];
  int i = threadIdx.x; // 512 threads
  float acc = 0.f;
  for (int o = 0; o < H2_; ++o) acc += W2[o] * W1[o * H2_ + i];
  v[i] = acc;
  __syncthreads();
  if (i < H_) { v1p[i] = v[i] + v[i + H_]; v2v[i] = v[i + H_]; }
  for (int idx = i; idx < 16 * H_; idx += H2_) {
    int n = idx >> 8;       // column 0..15
    int k = idx & (H_ - 1); // k 0..255
    float val = 0.f;
    if (n == 0) val = v[k] + v[k + H_];
    else if (n == 1) val = v[k + H_];
    bmatT[idx] = val;
  }
  if (i == 0) {
    float c = b2[0];
    for (int o = 0; o < H2_; ++o) c += W2[o] * b1[o];
    cb[0] = c;
  }
}

// ------------------------- per-token dots via WMMA: p = x·(v1+v2), d = x·v2,
// plus row abs-sum (for fpp). One wave per 16-token tile, K=256 over 64 WMMAs.
// Branch-free operands + 1-stage software pipeline.
__global__ void k_dots(const float* __restrict__ x, const float* __restrict__ v1p,
                       const float* __restrict__ v2v, const float* __restrict__ bmatT,
                       float* __restrict__ p, float* __restrict__ d,
                       float* __restrict__ ra) {
  int wave = (blockIdx.x * blockDim.x + threadIdx.x) >> 5;
  int lane = threadIdx.x & 31;
  const int tilesPerBatch = T_ / 16; // 128
  int b = wave / tilesPerBatch;
  int t0 = (wave % tilesPerBatch) * 16;
  int half = lane >> 4;
  int r = lane & 15;
  const float* xrow = x + (long)(b * T_ + t0 + r) * H_ + 2 * half;
  const float* brow = bmatT + r * H_ + 2 * half;
  float abss = 0.f;
#ifdef HAVE_WMMA_F32
  v8f acc = {};
  // A 16x4 f32: lanes 0-15 hold (M=r, K=k..k+1), lanes 16-31 (M=r, K=k+2..k+3)
  // B 4x16 f32: VGPR0 = row K=k+2*half, VGPR1 = K=k+2*half+1, N = r
  v2f a0 = *(const v2f*)(xrow);
  v2f b0 = *(const v2f*)(brow);
#pragma unroll 4
  for (int k = 0; k < H_; k += 4) {
    v2f a1, b1v;
    if (k + 4 < H_) {
      a1  = *(const v2f*)(xrow + k + 4);
      b1v = *(const v2f*)(brow + k + 4);
    }
    abss += fabsf(a0.x) + fabsf(a0.y);
    acc = __builtin_amdgcn_wmma_f32_16x16x4_f32(
        /*neg_a=*/false, a0, /*neg_b=*/false, b0,
        /*c_mod=*/(short)0, acc, /*reuse_a=*/false, /*reuse_b=*/false);
    a0 = a1;
    b0 = b1v;
  }
  abss += __shfl_xor(abss, 16);
  if (half == 0) ra[b * T_ + t0 + r] = abss;
  // D 16x16 f32: VGPR m -> (M=m+8*half, N=lane&15). Extract N=0 (p), N=1 (d).
  if (r < 2) {
#pragma unroll
    for (int m = 0; m < 8; ++m) {
      float val = acc[m];
      int row = m + 8 * half;
      if (r == 0) p[b * T_ + t0 + row] = val;
      else        d[b * T_ + t0 + row] = val;
    }
  }
#else
  // VALU fallback: each half-wave handles half of K for row r, combine via shfl
  const float* xr = x + (long)(b * T_ + t0 + r) * H_;
  float pa = 0.f, da = 0.f;
  int kbase = half * (H_ / 2);
  for (int k = 0; k < H_ / 2; ++k) {
    float xv = xr[kbase + k];
    pa += xv * v1p[kbase + k];
    da += xv * v2v[kbase + k];
    abss += fabsf(xv);
  }
  pa += __shfl_xor(pa, 16);
  da += __shfl_xor(da, 16);
  abss += __shfl_xor(abss, 16);
  if (half == 0) {
    p[b * T_ + t0 + r] = pa;
    d[b * T_ + t0 + r] = da;
    ra[b * T_ + t0 + r] = abss;
  }
#endif
}

__device__ __forceinline__ float blockReduceSum(float v, float* red) {
  int tid = threadIdx.x;
  red[tid] = v;
  __syncthreads();
  for (int s = 128; s > 0; s >>= 1) {
    if (tid < s) red[tid] += red[tid + s];
    __syncthreads();
  }
  float r = red[0];
  __syncthreads();
  return r;
}

// ---------------- window combine: w = clip(relu(p - windowSum(d)/cnt + c)),
// ori_weight out, sum_a out, fpp, then scaled+masked wz for the scan.
__global__ void k_window(const float* __restrict__ pin, const float* __restrict__ din,
                         const float* __restrict__ ra, const float* __restrict__ mask,
                         const float* __restrict__ cb, const int* __restrict__ is_training,
                         float* __restrict__ ori_w, float* __restrict__ suma_out,
                         float* __restrict__ wz, int* __restrict__ fpp_out) {
  __shared__ float red[256];
  __shared__ float sh_scale;
  __shared__ int sh_fpp;
  int b = blockIdx.x;
  int tid = threadIdx.x;
  float cnt_nz = 0.f, msum = 0.f;
#pragma unroll
  for (int i = 0; i < 8; ++i) {
    int t = tid + i * 256;
    cnt_nz += (ra[b * T_ + t] != 0.f) ? 1.f : 0.f;
    msum += mask[b * T_ + t];
  }
  float fppf = blockReduceSum(cnt_nz, red);
  float N = blockReduceSum(msum, red) - 1.f;
  float c = cb[0];
  float wloc[8];
  float wsum = 0.f;
#pragma unroll
  for (int i = 0; i < 8; ++i) {
    int t = tid + i * 256;
    float w;
    if (t == 0) {
      w = pin[b * T_] - din[b * T_] + c; // hist(t=0) = x0 -> hist·v2 = d0
    } else {
      int m = t < MAXH ? t : MAXH;
      float D = 0.f;
      for (int s = t - m; s < t; ++s) D += din[b * T_ + s];
      w = pin[b * T_ + t] - D / (float)m + c;
    }
    w = fminf(fmaxf(w, 0.f), RELU_MAX_);
    wloc[i] = w;
    ori_w[b * T_ + t] = w;
    wsum += w;
  }
  float suma = blockReduceSum(wsum, red);
  if (tid == 0) {
    suma_out[b] = suma;
    float scale = 1.f;
    if (is_training[0] != 0) {
      scale = N / fmaxf(suma, 1e-8f);
      if (suma == 0.f) scale = 0.f;
    }
    sh_scale = scale;
    sh_fpp = (int)fppf;
    fpp_out[b] = (int)fppf;
  }
  __syncthreads();
  float scale = sh_scale;
  int fpp = sh_fpp;
#pragma unroll
  for (int i = 0; i < 8; ++i) {
    int t = tid + i * 256;
    float w = wloc[i] * scale;
    wz[b * T_ + t] = (t > fpp) ? 0.f : w;
  }
}

// ---------- scalar fire recurrence: one wave per batch, lane-broadcast serial.
// Produces fired_flag, fire positions, compacted destination indices.
__global__ void k_flagscan(const float* __restrict__ wz, const int* __restrict__ fpp,
                           float* __restrict__ flag_out, int* __restrict__ fire_pos,
                           int* __restrict__ dest_idx, int* __restrict__ nfires) {
  int b = blockIdx.x;
  int lane = threadIdx.x;
  int f = fpp[b];
  float accw = 0.f;
  int nf = 0, emit = 0;
  for (int t0 = 0; t0 < T_; t0 += 32) {
    float wv = wz[b * T_ + t0 + lane];
    for (int j = 0; j < 32; ++j) {
      float wj = __shfl(wv, j);
      int t = t0 + j;
      bool fired = (accw + wj) > THRESH;
      accw = fired ? (wj - (THRESH - accw)) : (accw + wj);
      if (lane == j) flag_out[b * T_ + t] = fired ? 1.f : 0.f;
      if (fired) {
        if (lane == 0) {
          fire_pos[b * (T_ + 1) + nf] = t;
          dest_idx[b * (T_ + 1) + nf] = (t > f) ? -1 : emit;
        }
        if (t <= f) emit++;
        nf++;
      }
    }
  }
  if (lane == 0) {
    dest_idx[b * (T_ + 1) + nf] = emit; // last_s row always appended
    nfires[b] = nf;
  }
}

// ------- parallel segment reconstruction: one wave per emitted row.
// row_j = x[anchor] + sum_{t in (anchor, g_j]} (1 - wz_t) * x_t, then L2-normalize.
__global__ void k_emit(const float* __restrict__ x, const float* __restrict__ wz,
                       const int* __restrict__ fire_pos, const int* __restrict__ dest_idx,
                       const int* __restrict__ nfires, float* __restrict__ cif,
                       int* __restrict__ not_pad) {
  int b = blockIdx.x;
  int wave = threadIdx.x >> 5;
  int lane = threadIdx.x & 31;
  int j = blockIdx.y * 8 + wave;
  int nf = nfires[b];
  if (j > nf) return;
  int A, S;
  if (j == 0) { A = 0; S = 0; }
  else { A = fire_pos[b * (T_ + 1) + j - 1]; S = A + 1; }
  int E = (j < nf) ? fire_pos[b * (T_ + 1) + j] : (T_ - 1);
  const float* xb = x + (long)b * T_ * H_;
  float acc[8];
#pragma unroll
  for (int i = 0; i < 8; ++i) acc[i] = xb[(long)A * H_ + lane + 32 * i];
  for (int t = S; t <= E; ++t) {
    float coef = 1.f - wz[b * T_ + t];
#pragma unroll
    for (int i = 0; i < 8; ++i) acc[i] += coef * xb[(long)t * H_ + lane + 32 * i];
  }
  float ss = 0.f;
#pragma unroll
  for (int i = 0; i < 8; ++i) ss += acc[i] * acc[i];
  for (int m = 16; m >= 1; m >>= 1) ss += __shfl_xor(ss, m);
  int dest = dest_idx[b * (T_ + 1) + j];
  if (dest < 0) return;
  float inv = NORM_SCALAR_ / fmaxf(sqrtf(ss), 1e-12f);
  float* crow = cif + ((long)b * (T_ + 1) + dest) * H_;
#pragma unroll
  for (int i = 0; i < 8; ++i) crow[lane + 32 * i] = acc[i] * inv;
  if (lane == 0) not_pad[b * (T_ + 1) + dest] = 1;
}

extern "C" void kernel_launch(void* const* d_in, const int* in_sizes, int n_in,
                              void* d_out, int out_size, void* d_ws, size_t ws_size,
                              hipStream_t stream) {
  const float* x    = (const float*)d_in[0];
  const float* mask = (const float*)d_in[1];
  const float* W1   = (const float*)d_in[2];
  const float* b1   = (const float*)d_in[3];
  const float* W2   = (const float*)d_in[4];
  const float* b2   = (const float*)d_in[5];
  const int* is_tr  = (const int*)d_in[6];

  const long n_cif = (long)B_ * (T_ + 1) * H_;
  const long n_w   = (long)B_ * T_;
  const long n_np  = (long)B_ * (T_ + 1);

  float* out_cif  = (float*)d_out;
  float* out_ori  = out_cif + n_cif;
  float* out_flag = out_ori + n_w;
  int*   out_np   = (int*)(out_flag + n_w);
  float* out_suma = (float*)(out_np + n_np);

  float* ws    = (float*)d_ws;
  float* v1p   = ws;            // 256
  float* v2v   = ws + 256;      // 256
  float* cb    = ws + 512;      // 16
  float* bmatT = ws + 1024;     // 16*256 = 4096
  float* pA    = ws + 1024 + 4096;
  float* dA    = pA + n_w;
  float* raA   = dA + n_w;
  float* wzA   = raA + n_w;
  int* iws        = (int*)(wzA + n_w);
  int* fire_pos   = iws;
  int* dest_idx   = fire_pos + n_np;
  int* nfires     = dest_idx + n_np;
  int* fppA       = nfires + B_;

  k_zero<<<2048, 256, 0, stream>>>((float*)d_out, (long)out_size);
  k_fold<<<1, 512, 0, stream>>>(W1, b1, W2, b2, v1p, v2v, cb, bmatT);
  k_dots<<<(B_ * T_ / 16) / 8, 256, 0, stream>>>(x, v1p, v2v, bmatT, pA, dA, raA);
  k_window<<<B_, 256, 0, stream>>>(pA, dA, raA, mask, cb, is_tr,
                                   out_ori, out_suma, wzA, fppA);
  k_flagscan<<<B_, 32, 0, stream>>>(wzA, fppA, out_flag, fire_pos, dest_idx, nfires);
  k_emit<<<dim3(B_, (T_ + 1 + 7) / 8), 256, 0, stream>>>(x, wzA, fire_pos, dest_idx,
                                                         nfires, out_cif, out_np);
}